// Matching_82755429859695
// MI455X (gfx1250) — compile-verified
//
#include <hip/hip_runtime.h>
#include <hip/hip_bf16.h>

typedef float v2f __attribute__((ext_vector_type(2)));
typedef float v8f __attribute__((ext_vector_type(8)));
typedef float f4  __attribute__((ext_vector_type(4)));

#define Bsz 16
#define Cch 512
#define NPIX (16 * 64 * 64)   // B*H*W = 65536 pixels

// ---------------------------------------------------------------------------
// Kernel 1: q1[16,512] = style[16,512] @ W[512,512] via V_WMMA_F32_16X16X4_F32
// One wave (32 threads) computes one 16x16 tile of q1 (M = 16 = batch dim).
// A-frag layout (32-bit A 16x4): lanes 0-15 -> M=lane, {v0,v1}={K0,K1};
//                                lanes 16-31 -> M=lane-16, {v0,v1}={K2,K3}.
// B/C/D row-striped across lanes (B 4x16, C/D 16x16 per ISA 7.12.2).
// ---------------------------------------------------------------------------
__global__ void __launch_bounds__(32)
q1_gemm_wmma(const float* __restrict__ S,   // [16,512] style_coding
             const float* __restrict__ W,   // [512,512] weight
             float* __restrict__ Q)         // [16,512] q1 out
{
    const int lane = threadIdx.x & 31;
    const bool hi  = lane >= 16;
    const int  l   = lane & 15;
    const int  n0  = blockIdx.x * 16;

    v8f acc = {};
    for (int k0 = 0; k0 < Cch; k0 += 4) {
        const int ka = k0 + (hi ? 2 : 0);
        v2f a, b;
        a.x = S[l * Cch + ka];
        a.y = S[l * Cch + ka + 1];
        b.x = W[ka * Cch + n0 + l];
        b.y = W[(ka + 1) * Cch + n0 + l];
        // 8 args: (neg_a, A, neg_b, B, c_mod, C, reuse_a, reuse_b)
        acc = __builtin_amdgcn_wmma_f32_16x16x4_f32(
            false, a, false, b, (short)0, acc, false, false);
    }

#pragma unroll
    for (int v = 0; v < 8; ++v) {
        const int m = v + (hi ? 8 : 0);
        Q[m * Cch + n0 + l] = acc[v];
    }
}

// ---------------------------------------------------------------------------
// Kernel 2: per-pixel channel softmax, HBM-streaming bound.
// One wave32 per pixel; each lane owns 16 channels as 4x float4.
// Non-temporal 128-bit load of inputs / store of output (touched once);
// q1 row (2 KB, shared by 4096 pixels) uses regular cached loads.
// ---------------------------------------------------------------------------
__global__ void __launch_bounds__(256)
style_softmax(const float* __restrict__ x,   // [B,H,W,C]
              const float* __restrict__ q,   // [B,C] from kernel 1
              float* __restrict__ out)       // [B,H,W,C]
{
    const int wave = (blockIdx.x * blockDim.x + threadIdx.x) >> 5;
    const int lane = threadIdx.x & 31;
    if (wave >= NPIX) return;

    const long long base = (long long)wave * Cch;
    const int b = wave >> 12;                       // 4096 pixels per batch

    const f4* __restrict__ xp = (const f4*)(x + base);
    const f4* __restrict__ qp = (const f4*)(q + (long long)b * Cch);
    f4* __restrict__ op       = (f4*)(out + base);

    f4 xv[4], sv[4];
#pragma unroll
    for (int j = 0; j < 4; ++j) {
        const int idx = j * 32 + lane;              // f4 units: c = j*128 + lane*4
        xv[j] = __builtin_nontemporal_load(&xp[idx]);
        f4 qv = qp[idx];
        sv[j] = qv * xv[j];
    }

    // ---- max over 512 channels (16 local + 5 shuffle steps, wave32) ----
    float m = sv[0].x;
#pragma unroll
    for (int j = 0; j < 4; ++j) {
        m = fmaxf(m, fmaxf(fmaxf(sv[j].x, sv[j].y), fmaxf(sv[j].z, sv[j].w)));
    }
#pragma unroll
    for (int off = 16; off > 0; off >>= 1)
        m = fmaxf(m, __shfl_xor(m, off, 32));

    // ---- exp and sum ----
    f4 ev[4];
    float s = 0.0f;
#pragma unroll
    for (int j = 0; j < 4; ++j) {
        ev[j].x = __expf(sv[j].x - m);
        ev[j].y = __expf(sv[j].y - m);
        ev[j].z = __expf(sv[j].z - m);
        ev[j].w = __expf(sv[j].w - m);
        s += (ev[j].x + ev[j].y) + (ev[j].z + ev[j].w);
    }
#pragma unroll
    for (int off = 16; off > 0; off >>= 1)
        s += __shfl_xor(s, off, 32);

    const float inv = 1.0f / s;

    // ---- o = softmax * v  (v == x), non-temporal 128-bit stores ----
#pragma unroll
    for (int j = 0; j < 4; ++j) {
        f4 o = ev[j] * xv[j] * inv;
        __builtin_nontemporal_store(o, &op[j * 32 + lane]);
    }
}

extern "C" void kernel_launch(void* const* d_in, const int* in_sizes, int n_in,
                              void* d_out, int out_size, void* d_ws, size_t ws_size,
                              hipStream_t stream) {
    const float* inputs = (const float*)d_in[0];   // [16,64,64,512]
    const float* style  = (const float*)d_in[1];   // [16,512]
    const float* weight = (const float*)d_in[2];   // [512,512]
    float* out = (float*)d_out;
    float* q1  = (float*)d_ws;                     // 16*512*4 = 32 KB scratch

    // 32 N-tiles of 16, one wave each
    q1_gemm_wmma<<<Cch / 16, 32, 0, stream>>>(style, weight, q1);

    // 65536 pixels, 8 waves (pixels) per 256-thread block
    style_softmax<<<NPIX / 8, 256, 0, stream>>>(inputs, q1, out);
}